// AtomicDifferentiatedDense_68315749810418
// MI455X (gfx1250) — compile-verified
//
#include <hip/hip_runtime.h>

typedef __attribute__((ext_vector_type(2))) float v2f;
typedef __attribute__((ext_vector_type(8))) float v8f;

#define K_DIM 128
#define N_DIM 128
#define WAVES 4
#define ROWS_PER_WAVE 32
#define ROWS_PER_BLOCK (WAVES * ROWS_PER_WAVE)   // 128
#define XROW_STRIDE 132                          // dword stride: conflict-free A reads

// ---------------- kernel 1: zero per-expert counters ----------------
__global__ void add_init(int* __restrict__ cnt, int E) {
    if ((int)threadIdx.x < E) cnt[threadIdx.x] = 0;
}

// ---------------- kernel 2: bin rows by matching expert ----------------
__global__ void add_classify(const int* __restrict__ atom_numbers,
                             const int* __restrict__ cases,
                             int* __restrict__ cnt,
                             int* __restrict__ idxlist,
                             int M, int E) {
    int r = blockIdx.x * blockDim.x + threadIdx.x;
    if (r >= M) return;
    int an = atom_numbers[r];
    for (int e = 0; e < E; ++e) {
        if (an == cases[e]) {
            int pos = atomicAdd(&cnt[e], 1);
            idxlist[(size_t)e * M + pos] = r;
            return;
        }
    }
}

// ---------------- kernel 3: zero output rows with no matching expert ----------------
__global__ void zero_unmatched(const int* __restrict__ atom_numbers,
                               const int* __restrict__ cases,
                               float* __restrict__ out,
                               int M, int E) {
    int row  = (int)((blockIdx.x * blockDim.x + threadIdx.x) >> 5);
    int lane = threadIdx.x & 31;
    if (row >= M) return;
    int an = atom_numbers[row];
    bool matched = false;
    for (int e = 0; e < E; ++e) matched = matched || (an == cases[e]);
    if (!matched) {
        float4 z = make_float4(0.f, 0.f, 0.f, 0.f);
        *(float4*)(out + (size_t)row * N_DIM + (lane << 2)) = z;
    }
}

// ---------------- kernel 4: grouped gather-GEMM with f32 WMMA ----------------
// block = 128 threads (4 waves), handles 128 gathered rows of ONE expert.
// Each wave: 32 rows (2 M-tiles) x 128 cols, K=128 in 32 WMMA K-steps,
// with A/B fragments software-pipelined one K-step ahead (ping-pong regs).
// LDS: W[e] swizzled into B-fragment order (16384 floats) + 4 x-tiles of
//      32 rows x stride-132 (conflict-free b64 A-fragment reads).
__global__ __launch_bounds__(128) void moe_gemm(
        const float* __restrict__ x,
        const float* __restrict__ W,
        const float* __restrict__ bias,
        const int* __restrict__ cnt,
        const int* __restrict__ idxlist,
        float* __restrict__ out,
        int M, int maxTiles) {
    extern __shared__ float lds[];

    const int e        = blockIdx.x / maxTiles;
    const int tile     = blockIdx.x % maxTiles;
    const int count    = cnt[e];
    const int rowStart = tile * ROWS_PER_BLOCK;
    if (rowStart >= count) return;   // block-uniform early exit (before barriers)

    const int tid  = threadIdx.x;
    const int lane = tid & 31;
    const int w    = tid >> 5;

    // --- stage W[e] (K=128 x N=128) into LDS in B-fragment order ---
    // fragment dword index = ((kk*8 + nn)*32 + lane)*2 + j
    // element (k,n): kk=k>>2, kr=k&3, nn=n>>4, nl=n&15, lane=(kr>=2?16:0)+nl, j=kr&1
    const float* We = W + (size_t)e * K_DIM * N_DIM;
    for (int g = tid; g < K_DIM * N_DIM; g += blockDim.x) {
        float val = We[g];
        int k  = g >> 7, n = g & 127;
        int kk = k >> 2, kr = k & 3;
        int nn = n >> 4, nl = n & 15;
        int fl = ((kr >> 1) << 4) | nl;
        int j  = kr & 1;
        lds[(((kk << 3) + nn) << 6) + (fl << 1) + j] = val;
    }

    // --- stage this wave's 32 gathered x rows (coalesced 512B per row) ---
    const int XB    = K_DIM * N_DIM;                       // 16384 floats
    const int xbase = XB + w * (ROWS_PER_WAVE * XROW_STRIDE);
    const int rbase = rowStart + w * ROWS_PER_WAVE;
    for (int i = 0; i < ROWS_PER_WAVE; ++i) {
        int gi = rbase + i;
        int ci = gi < count ? gi : count - 1;  // clamp (extra rows computed, not stored)
        int ri = idxlist[(size_t)e * M + ci];
        const float4 v = *(const float4*)(x + (size_t)ri * K_DIM + (lane << 2));
        *(float4*)(lds + xbase + i * XROW_STRIDE + (lane << 2)) = v;
    }
    __syncthreads();

    // --- 32x128 output stripe per wave ---
    v8f acc[2][8];
#pragma unroll
    for (int m = 0; m < 2; ++m)
#pragma unroll
        for (int nn = 0; nn < 8; ++nn) acc[m][nn] = (v8f)(0.f);

    // A fragment (ISA 7.12.2): lanes 0-15 -> M=lane, K={4kk,4kk+1};
    //                          lanes 16-31 -> M=lane-16, K={4kk+2,4kk+3}
    const int aoff0 = xbase + (lane & 15) * XROW_STRIDE + ((lane >> 4) << 1);
    const int aoff1 = aoff0 + 16 * XROW_STRIDE;
    const int boff  = lane << 1;

    // ping-pong fragment buffers, loaded one K-step ahead
    v2f a0[2], a1[2], bf[2][8];

    {   // prologue: load K-step 0 into buffer 0
        a0[0] = *(const v2f*)(lds + aoff0);
        a1[0] = *(const v2f*)(lds + aoff1);
#pragma unroll
        for (int nn = 0; nn < 8; ++nn)
            bf[0][nn] = *(const v2f*)(lds + boff + (nn << 6));
    }

#pragma unroll 2
    for (int kk = 0; kk < 32; ++kk) {
        const int p = kk & 1;
        const int q = p ^ 1;
        const int kn = (kk + 1) & 31;   // last iter wraps to 0: in-bounds, discarded
        // issue ALL loads for the NEXT K-step before consuming this one
        a0[q] = *(const v2f*)(lds + aoff0 + (kn << 2));
        a1[q] = *(const v2f*)(lds + aoff1 + (kn << 2));
        const float* bp = lds + (kn << 9) + boff;
#pragma unroll
        for (int nn = 0; nn < 8; ++nn) bf[q][nn] = *(const v2f*)(bp + (nn << 6));

#pragma unroll
        for (int nn = 0; nn < 8; ++nn)
            acc[0][nn] = __builtin_amdgcn_wmma_f32_16x16x4_f32(
                false, a0[p], false, bf[p][nn], (short)0, acc[0][nn], false, false);
#pragma unroll
        for (int nn = 0; nn < 8; ++nn)
            acc[1][nn] = __builtin_amdgcn_wmma_f32_16x16x4_f32(
                false, a1[p], false, bf[p][nn], (short)0, acc[1][nn], false, false);
    }

    // --- epilogue: + bias, relu, scatter rows back ---
    // C/D layout: VGPR v, lanes 0-15: M=v, N=lane; lanes 16-31: M=v+8, N=lane-16
    const float* be = bias + (size_t)e * N_DIM;
    const int nl = lane & 15;
    const int hi = lane >> 4;
    float bv[8];
#pragma unroll
    for (int nn = 0; nn < 8; ++nn) bv[nn] = be[(nn << 4) + nl];

#pragma unroll
    for (int m = 0; m < 2; ++m) {
#pragma unroll
        for (int v = 0; v < 8; ++v) {
            int gi = rbase + m * 16 + v + (hi << 3);
            if (gi < count) {
                int ri = idxlist[(size_t)e * M + gi];
                float* orow = out + (size_t)ri * N_DIM;
#pragma unroll
                for (int nn = 0; nn < 8; ++nn) {
                    float val = acc[m][nn][v] + bv[nn];
                    orow[(nn << 4) + nl] = fmaxf(val, 0.f);
                }
            }
        }
    }
}

// ---------------- launcher ----------------
extern "C" void kernel_launch(void* const* d_in, const int* in_sizes, int n_in,
                              void* d_out, int out_size, void* d_ws, size_t ws_size,
                              hipStream_t stream) {
    const float* x            = (const float*)d_in[0];
    const int*   atom_numbers = (const int*)d_in[1];
    const float* W            = (const float*)d_in[2];
    const float* b            = (const float*)d_in[3];
    const int*   cases        = (const int*)d_in[4];
    float*       out          = (float*)d_out;

    const int M = in_sizes[1];          // B * A rows
    const int E = in_sizes[4];          // number of experts (4)

    int* cnt     = (int*)d_ws;                    // E counters
    int* idxlist = (int*)((char*)d_ws + 256);     // E * M row indices

    hipLaunchKernelGGL(add_init, dim3(1), dim3(32), 0, stream, cnt, E);
    hipLaunchKernelGGL(add_classify, dim3((M + 255) / 256), dim3(256), 0, stream,
                       atom_numbers, cases, cnt, idxlist, M, E);
    hipLaunchKernelGGL(zero_unmatched, dim3((M * 32 + 255) / 256), dim3(256), 0, stream,
                       atom_numbers, cases, out, M, E);

    const int maxTiles = (M + ROWS_PER_BLOCK - 1) / ROWS_PER_BLOCK;
    const size_t shbytes =
        (size_t)(K_DIM * N_DIM + WAVES * ROWS_PER_WAVE * XROW_STRIDE) * sizeof(float);
    hipLaunchKernelGGL(moe_gemm, dim3(E * maxTiles), dim3(128), shbytes, stream,
                       x, W, b, cnt, idxlist, out, M, maxTiles);
}